// OsuRatingSystem_78116865180217
// MI455X (gfx1250) — compile-verified
//
#include <hip/hip_runtime.h>

// out[b] = dot(player_table[pidx[b], :64], map_table[midx[b], :64])
//
// Memory-bound gather kernel for MI455X (gfx1250, wave32).
//  - Each wave handles 2 batch elements/iter: lanes 0-15 -> e0, lanes 16-31 -> e1.
//  - Each lane loads one float4 (16B): a 16-lane group fetches the full 256B
//    embedding row contiguously (global_load_b128, cacheline-exact).
//  - Dot4 per lane, then 4-step __shfl_xor butterfly within each 16-lane half.
//  - Software-pipelined indices: next iteration's pidx/midx are loaded once,
//    used to issue global_prefetch_b8 on the gathered rows, then carried in
//    registers to become the current indices next iteration. This gives a
//    full-iteration prefetch distance and halves index traffic.
//
// Roofline: ~256MB player rows (random over a 256MB table -> HBM) + ~51MB map
// rows (200K x 256B table fits in 192MB L2) + 12MB indices/output
// => ~320MB HBM => ~14us floor at 23.3 TB/s. Compute (0.128 GFLOP) is noise;
// WMMA is inapplicable (no matrix structure; diag-of-GEMM wastes 16x + shuffles).

__global__ __launch_bounds__(256)
void osu_rating_dot_kernel(const int* __restrict__ pidx,
                           const int* __restrict__ midx,
                           const float* __restrict__ ptab,
                           const float* __restrict__ mtab,
                           float* __restrict__ out,
                           int n)
{
    const int tid    = blockIdx.x * blockDim.x + threadIdx.x;
    const int lane   = threadIdx.x & 31;   // wave32
    const int group  = lane >> 4;          // which of the 2 elements this half-wave owns
    const int gl     = lane & 15;          // lane within 16-lane group
    const int waveId = tid >> 5;
    const int nWaves = (gridDim.x * blockDim.x) >> 5;
    const int stride = nWaves * 2;         // elements consumed per grid iteration

    int e = waveId * 2 + group;
    if (e >= n) return;

    // Prologue: load this element's indices once; thereafter indices rotate
    // through registers (loaded exactly once, at prefetch time).
    int pi = pidx[e];
    int mi = midx[e];

    for (; e < n; e += stride) {
        // ---- issue prefetch for next iteration's rows, carry its indices ----
        const int en = e + stride;
        int pin = 0, min_ = 0;
        if (en < n) {
            pin  = pidx[en];
            min_ = midx[en];
            // 16 lanes x 16B spread covers every cacheline of the 256B row.
            __builtin_prefetch(ptab + (size_t)pin  * 64 + (gl << 2), 0, 1);
            __builtin_prefetch(mtab + (size_t)min_ * 64 + (gl << 2), 0, 1);
        }

        // ---- gather this iteration's rows (cooperative, contiguous 256B) ----
        const float4* __restrict__ prow = (const float4*)(ptab + (size_t)pi * 64);
        const float4* __restrict__ mrow = (const float4*)(mtab + (size_t)mi * 64);

        const float4 p = prow[gl];   // global_load_b128
        const float4 m = mrow[gl];   // global_load_b128

        float s = fmaf(p.x, m.x, fmaf(p.y, m.y, fmaf(p.z, m.z, p.w * m.w)));

        // ---- butterfly reduction within each 16-lane half (wave32-safe) ----
        s += __shfl_xor(s, 8, 32);
        s += __shfl_xor(s, 4, 32);
        s += __shfl_xor(s, 2, 32);
        s += __shfl_xor(s, 1, 32);

        if (gl == 0) {
            out[e] = s;
        }

        // rotate pipelined indices into place for the next iteration
        pi = pin;
        mi = min_;
    }
}

extern "C" void kernel_launch(void* const* d_in, const int* in_sizes, int n_in,
                              void* d_out, int out_size, void* d_ws, size_t ws_size,
                              hipStream_t stream)
{
    // setup_inputs() order: player_indices, map_indices, player_table, map_table
    const int*   pidx = (const int*)d_in[0];
    const int*   midx = (const int*)d_in[1];
    const float* ptab = (const float*)d_in[2];
    const float* mtab = (const float*)d_in[3];
    float*       out  = (float*)d_out;
    const int    n    = in_sizes[0];

    const int block = 256;                        // 8 wave32 waves per block
    const int elemsPerBlock = (block / 32) * 2;   // 16 outputs / block / iter
    int grid = (n + elemsPerBlock - 1) / elemsPerBlock;
    if (grid > 16384) grid = 16384;               // grid-stride + prefetch distance
    if (grid < 1) grid = 1;

    osu_rating_dot_kernel<<<grid, block, 0, stream>>>(pidx, midx, ptab, mtab, out, n);
}